// DotProductDecoder_77275051589763
// MI455X (gfx1250) — compile-verified
//
#include <hip/hip_runtime.h>
#include <math.h>

// DotProductDecoder for MI455X (gfx1250, wave32).
//
// out[e] = sigmoid( dot( h[src[e], :256], h[dst[e], :256] ) )
//
// Strategy: one wave32 handles 16 edges. A = 16 gathered src rows (16xK),
// B = 16 gathered dst rows laid out as columns (Kx16). Chain 64x
// V_WMMA_F32_16X16X4_F32 over K=256; diag(A x B) = the 16 dot products,
// accumulated in fp32 by the matrix unit (no cross-lane reduction needed).
// Workload is L2-gather-bound (h table = 51.2 MB < 192 MB L2), so the 16x
// redundant WMMA flops are free and we keep full fp32 precision.

typedef __attribute__((ext_vector_type(2))) float v2f;
typedef __attribute__((ext_vector_type(8))) float v8f;

__global__ __launch_bounds__(256) void DotProductDecoder_wmma_kernel(
    const float* __restrict__ h,          // [N, 256] fp32
    const long long* __restrict__ eidx,   // [2, E] int64 (row0=src, row1=dst)
    float* __restrict__ out,              // [E] fp32
    int E)
{
    const int lane      = threadIdx.x & 31;
    const int waveInBlk = threadIdx.x >> 5;
    const long long group = (long long)blockIdx.x * 8 + waveInBlk;
    const long long base  = group * 16;          // first edge of this wave's group
    if (base >= E) return;                       // whole wave inactive -> no WMMA

    const int sub = lane & 15;                   // edge slot within the group
    const int hi  = lane >> 4;                   // half-wave: K-offset selector

    long long e = base + sub;
    if (e >= E) e = E - 1;                       // clamp: keeps EXEC all-ones

    const long long s = eidx[e];                 // src node id
    const long long d = eidx[(long long)E + e];  // dst node id

    // 32-bit A 16x4 layout: lane m holds A[m][k0+0..1], lane m+16 holds A[m][k0+2..3].
    // 32-bit B 4x16 layout mirrors it for the gathered dst rows.
    const float* __restrict__ pA = h + s * 256 + 2 * hi;
    const float* __restrict__ pB = h + d * 256 + 2 * hi;

    v8f c = {};   // 16x16 fp32 accumulator (8 VGPRs)
#pragma unroll 8
    for (int k = 0; k < 256; k += 4) {
        v2f a = *(const v2f*)(pA + k);           // global_load_b64
        v2f b = *(const v2f*)(pB + k);           // global_load_b64
        // (neg_a, A, neg_b, B, c_mod, C, reuse_a, reuse_b)
        c = __builtin_amdgcn_wmma_f32_16x16x4_f32(
                false, a, false, b, (short)0, c, false, false);
    }

    // Diagonal of the 16x16 result:
    //   edge m in [0,8)  -> VGPR m     on lane m
    //   edge m in [8,16) -> VGPR m-8   on lane m+16  (lanes 24..31)
    int pick = -1;
    long long eo = -1;
    if (lane < 8)        { pick = lane;      eo = base + lane;        }
    else if (lane >= 24) { pick = lane - 24; eo = base + (lane - 16); }

    float v = 0.0f;
#pragma unroll
    for (int i = 0; i < 8; ++i) v = (pick == i) ? c[i] : v;

    if (pick >= 0 && eo < (long long)E) {
        out[eo] = 1.0f / (1.0f + __expf(-v));    // sigmoid
    }
}

extern "C" void kernel_launch(void* const* d_in, const int* in_sizes, int n_in,
                              void* d_out, int out_size, void* d_ws, size_t ws_size,
                              hipStream_t stream) {
    const float*     h    = (const float*)d_in[0];
    const long long* eidx = (const long long*)d_in[1];
    float*           out  = (float*)d_out;

    const int E = in_sizes[1] / 2;               // edge_index is [2, E]

    const int groups = (E + 15) / 16;            // one wave per 16 edges
    const int wavesPerBlock = 8;                 // 256 threads = 8 wave32
    const int blocks = (groups + wavesPerBlock - 1) / wavesPerBlock;

    DotProductDecoder_wmma_kernel<<<blocks, 256, 0, stream>>>(h, eidx, out, E);
}